// ExtractTsFeatures_31688268710448
// MI455X (gfx1250) — compile-verified
//
#include <hip/hip_runtime.h>

#define TLEN  2048
#define FDIM  16
#define BDIM  512
#define NTH   256
#define NOUT  3146
#define NBINS 1025

typedef float v2f __attribute__((ext_vector_type(2)));
typedef float v8f __attribute__((ext_vector_type(8)));
typedef unsigned int u32x4 __attribute__((ext_vector_type(4)));
typedef int i32x4 __attribute__((ext_vector_type(4)));
typedef int i32x8 __attribute__((ext_vector_type(8)));

#if defined(__has_builtin)
#if __has_builtin(__builtin_amdgcn_tensor_load_to_lds) && __has_builtin(__builtin_amdgcn_s_wait_tensorcnt)
#define USE_TDM 1
#endif
#endif
#ifndef USE_TDM
#define USE_TDM 0
#endif

__device__ __forceinline__ float sgnf(float v) {
    return (v > 0.f) ? 1.f : ((v < 0.f) ? -1.f : 0.f);
}

__device__ float blk_sum(float v, float* red) {
    int tid = threadIdx.x;
    red[tid] = v; __syncthreads();
    #pragma unroll
    for (int s = NTH / 2; s > 0; s >>= 1) {
        if (tid < s) red[tid] += red[tid + s];
        __syncthreads();
    }
    float r = red[0]; __syncthreads();
    return r;
}

__device__ float blk_min(float v, float* red) {
    int tid = threadIdx.x;
    red[tid] = v; __syncthreads();
    #pragma unroll
    for (int s = NTH / 2; s > 0; s >>= 1) {
        if (tid < s) red[tid] = fminf(red[tid], red[tid + s]);
        __syncthreads();
    }
    float r = red[0]; __syncthreads();
    return r;
}

__device__ float blk_max(float v, float* red) {
    int tid = threadIdx.x;
    red[tid] = v; __syncthreads();
    #pragma unroll
    for (int s = NTH / 2; s > 0; s >>= 1) {
        if (tid < s) red[tid] = fmaxf(red[tid], red[tid + s]);
        __syncthreads();
    }
    float r = red[0]; __syncthreads();
    return r;
}

__global__ __launch_bounds__(NTH)
void ts_features_kernel(const float* __restrict__ x, float* __restrict__ out) {
    __shared__ float  colx[TLEN];        // raw column, persists whole kernel
    __shared__ float2 bufA[TLEN];        // FFT ping
    __shared__ float2 bufB[TLEN];        // FFT pong (aliased as sort buffer first)
    __shared__ float  ampb[NBINS];
    __shared__ float  red[NTH];
    __shared__ float  ac_sh[9];

    const int col = blockIdx.x;
    const int b = col >> 4, f = col & 15;
    const float* xp = x + (size_t)b * TLEN * FDIM + f;
    float* op = out + (size_t)col * NOUT;
    const int tid = threadIdx.x;
    const float n = (float)TLEN;

    // ---- load column (stride FDIM floats) into LDS ----
#if USE_TDM
    // Tensor Data Mover: DMA the strided column straight into LDS.
    // Tile = 2048 rows x 1 element, row stride 16 elements, 4B data.
    if (tid < 32) {
        unsigned long long ga = (unsigned long long)(size_t)xp;     // tile start
        unsigned int ldsb = (unsigned int)(size_t)&colx[0];          // LDS byte offset (addr[31:0])
        u32x4 g0;
        g0[0] = 1u;                                   // count=1, is_restore=0, gather=0
        g0[1] = ldsb;                                 // lds_addr
        g0[2] = (unsigned int)ga;                     // global_addr[31:0]
        g0[3] = (unsigned int)((ga >> 32) & 0x01FFFFFFu) | 0x80000000u; // addr[56:32] | type=2
        i32x8 g1;
        g1[0] = (int)(2u << 16);      // workgroup_mask=0, data_size=2 (4 bytes)
        g1[1] = (int)(16u << 16);     // tensor_dim0 = 16 (low half -> bits[63:48])
        g1[2] = (int)(2048u << 16);   // tensor_dim0 hi=0, tensor_dim1 = 2048 (low half)
        g1[3] = (int)(1u << 16);      // tensor_dim1 hi=0, tile_dim0 = 1
        g1[4] = (int)2048;            // tile_dim1 = 2048, tile_dim2 = 0
        g1[5] = 16;                   // tensor_dim0_stride = 16 elements
        g1[6] = 0;                    // stride0 hi, tensor_dim1_stride lo
        g1[7] = 0;
        i32x4 g2; g2[0] = 0; g2[1] = 0; g2[2] = 0; g2[3] = 0;
        i32x4 g3; g3[0] = 0; g3[1] = 0; g3[2] = 0; g3[3] = 0;
#if __clang_major__ >= 23
        i32x8 g1b; g1b[0]=0; g1b[1]=0; g1b[2]=0; g1b[3]=0; g1b[4]=0; g1b[5]=0; g1b[6]=0; g1b[7]=0;
        __builtin_amdgcn_tensor_load_to_lds(g0, g1, g2, g3, g1b, 0);
#else
        __builtin_amdgcn_tensor_load_to_lds(g0, g1, g2, g3, 0);
#endif
        __builtin_amdgcn_s_wait_tensorcnt(0);
    }
    __syncthreads();
#else
    for (int t = tid; t < TLEN; t += NTH) colx[t] = xp[(size_t)t * FDIM];
    __syncthreads();
#endif

    // ---- pass A: raw moments, extremes, diffs, sign stats ----
    float s1 = 0.f, s2 = 0.f, cpos = 0.f, crx = 0.f, sd = 0.f, sad = 0.f, sdd = 0.f;
    float mn = __builtin_inff(), mx = -__builtin_inff(), ma = 0.f;
    for (int t = tid; t < TLEN; t += NTH) {
        float v = colx[t];
        s1 += v; s2 += v * v;
        mn = fminf(mn, v); mx = fmaxf(mx, v); ma = fmaxf(ma, fabsf(v));
        if (v > 0.f) cpos += 1.f;
        if (t < TLEN - 1) { if (sgnf(v) != sgnf(colx[t + 1])) crx += 1.f; }
        if (t < TLEN - 2) {
            float d = colx[t + 1] - colx[t + 2];   // xd = x[1:-1] - x[2:]
            sd += d; sad += fabsf(d); sdd += d * d;
        }
    }
    s1 = blk_sum(s1, red);  s2 = blk_sum(s2, red);
    mn = blk_min(mn, red);  mx = blk_max(mx, red);  ma = blk_max(ma, red);
    cpos = blk_sum(cpos, red);  crx = blk_sum(crx, red);
    sd = blk_sum(sd, red);  sad = blk_sum(sad, red);  sdd = blk_sum(sdd, red);

    const float mean = s1 / n;

    // ---- pass B: centered moments ----
    float sc2 = 0.f, sc3 = 0.f, sc4 = 0.f, ccpos = 0.f, ccrx = 0.f;
    for (int t = tid; t < TLEN; t += NTH) {
        float c = colx[t] - mean;
        float c2 = c * c;
        sc2 += c2; sc3 += c2 * c; sc4 += c2 * c2;
        if (c > 0.f) ccpos += 1.f;
        if (t < TLEN - 1) { if (sgnf(c) != sgnf(colx[t + 1] - mean)) ccrx += 1.f; }
    }
    sc2 = blk_sum(sc2, red); sc3 = blk_sum(sc3, red); sc4 = blk_sum(sc4, red);
    ccpos = blk_sum(ccpos, red); ccrx = blk_sum(ccrx, red);

    const float var  = sc2 / n;
    const float stdv = (var > 0.f) ? sqrtf(var) : 0.f;

    // ---- bitonic sort (in bufB storage) for quartiles ----
    float* sortf = (float*)bufB;
    for (int t = tid; t < TLEN; t += NTH) sortf[t] = colx[t];
    __syncthreads();
    for (int k = 2; k <= TLEN; k <<= 1) {
        for (int j = k >> 1; j > 0; j >>= 1) {
            for (int i = tid; i < TLEN; i += NTH) {
                int ixj = i ^ j;
                if (ixj > i) {
                    float a0 = sortf[i], b0 = sortf[ixj];
                    bool up = ((i & k) == 0);
                    if (up ? (a0 > b0) : (a0 < b0)) { sortf[i] = b0; sortf[ixj] = a0; }
                }
            }
            __syncthreads();
        }
    }
    const float q0 = sortf[512], q1 = sortf[1024], q2 = sortf[1536];
    const float tb0 = colx[0], tb1 = colx[512], tb2 = colx[1024],
                tb3 = colx[1536], tb4 = colx[2047];
    __syncthreads();

    // ---- pass C: counts/crossings vs 8 thresholds (5 tb + 3 q) ----
    float thr[8] = { tb0, tb1, tb2, tb3, tb4, q0, q1, q2 };
    float cntv[8], crov[8];
    for (int h = 0; h < 8; ++h) {
        float th = thr[h], cnt = 0.f, cro = 0.f;
        for (int t = tid; t < TLEN; t += NTH) {
            float v = colx[t] - th;
            if (v > 0.f) cnt += 1.f;
            if (t < TLEN - 1) { if (sgnf(v) != sgnf(colx[t + 1] - th)) cro += 1.f; }
        }
        cntv[h] = blk_sum(cnt, red);
        crov[h] = blk_sum(cro, red);
    }

    // ---- Stockham radix-2 DIF FFT (11 stages, natural order out) ----
    for (int t = tid; t < TLEN; t += NTH) { bufA[t].x = colx[t]; bufA[t].y = 0.f; }
    __syncthreads();
    float2* Ap = bufA; float2* Bp = bufB;
    int ms = 0;
    for (int l = TLEN / 2; l >= 1; l >>= 1) {
        int m = 1 << ms;
        for (int q = tid; q < TLEN / 2; q += NTH) {
            int j = q >> ms;
            int k = q & (m - 1);
            float2 c0 = Ap[k + j * m];
            float2 c1 = Ap[k + j * m + TLEN / 2];
            float angr = -3.14159265358979323846f * (float)j / (float)l;
            float sn, cs; sincosf(angr, &sn, &cs);
            float dx = c0.x - c1.x, dy = c0.y - c1.y;
            float2 e0; e0.x = c0.x + c1.x; e0.y = c0.y + c1.y;
            float2 e1; e1.x = cs * dx - sn * dy; e1.y = cs * dy + sn * dx;
            Bp[k + 2 * j * m]     = e0;
            Bp[k + 2 * j * m + m] = e1;
        }
        __syncthreads();
        float2* tp = Ap; Ap = Bp; Bp = tp;
        ++ms;
    }

    // ---- amp / angle outputs ----
    for (int k = tid; k < NBINS; k += NTH) {
        float re = Ap[k].x, im = Ap[k].y;
        float a = sqrtf(re * re + im * im);
        ampb[k] = a;
        op[53 + k]         = a;
        op[53 + NBINS + k] = atan2f(im, re);
    }
    __syncthreads();

    // ---- amp aggregate stats (n2 = 1025) ----
    float pa = 0.f;
    for (int k = tid; k < NBINS; k += NTH) pa += ampb[k];
    const float Sa = blk_sum(pa, red);
    const float n2 = (float)NBINS;
    const float am = Sa / n2;
    float a2 = 0.f, a3 = 0.f, a4 = 0.f;
    for (int k = tid; k < NBINS; k += NTH) {
        float c = ampb[k] - am, c2 = c * c;
        a2 += c2; a3 += c2 * c; a4 += c2 * c2;
    }
    a2 = blk_sum(a2, red); a3 = blk_sum(a3, red); a4 = blk_sum(a4, red);
    const float av   = a2 / n2;
    const float astd = (av > 0.f) ? sqrtf(av) : 0.f;
    const float ask  = (astd != 0.f) ? (n2 / ((n2 - 1.f) * (n2 - 2.f))) * a3 / (astd * astd * astd) : 0.f;
    const float alpha2 = n2 * (n2 + 1.f) * (n2 - 1.f) / ((n2 - 2.f) * (n2 - 3.f));
    const float beta2  = 3.f * (n2 - 1.f) * (n2 - 1.f) / ((n2 - 2.f) * (n2 - 3.f));
    const float aku  = ((a2 != 0.f) ? alpha2 * a4 / (a2 * a2) : 0.f) - beta2;

    // ratio = amp / sum(amp): stats are exact rescalings (S cancels in skew/kurt)
    const float invS = (Sa != 0.f) ? 1.f / Sa : 0.f;
    for (int k = tid; k < NBINS; k += NTH)
        op[53 + 2 * NBINS + k] = ampb[k] * invS;
    const float rm = am * invS;
    const float rv = av * invS * invS;
    const float rsk = ask, rku = aku;

    // ---- autocorr lags 1..9 via f32 WMMA Gram matrix (wave 0) ----
    __syncthreads();
    if (tid < 32) {
        int lane = tid, mrow = lane & 15, hh = lane >> 4;
        v8f acc = {0.f, 0.f, 0.f, 0.f, 0.f, 0.f, 0.f, 0.f};
        for (int t0 = 0; t0 < 2032; t0 += 4) {
            int base = t0 + 2 * hh + mrow;
            v2f a;
            a.x = colx[base]     - mean;   // A[m][k] = B[k][n] = xc[t0+k+idx]
            a.y = colx[base + 1] - mean;
            acc = __builtin_amdgcn_wmma_f32_16x16x4_f32(
                false, a, false, a, (short)0, acc, false, false);
        }
        if (hh == 0 && mrow >= 1 && mrow <= 9) {
            int l = mrow;
            float g = acc[0];                       // G[0][l]
            for (int t = 2032; t <= 2047 - l; ++t)  // tail terms
                g += (colx[t] - mean) * (colx[t + l] - mean);
            float acl = g / (float)(TLEN - l);
            ac_sh[l - 1] = (var != 0.f) ? acl / var : 0.f;
        }
    }
    __syncthreads();

    // ---- scalar outputs ----
    if (tid == 0) {
        op[0] = mean;
        op[1] = mn;
        op[2] = mx;
        op[3] = (s2 > 0.f) ? sqrtf(s2 / n) : 0.f;
        op[4] = var;
        op[5] = stdv;
        float coef = n / ((n - 1.f) * (n - 2.f));
        op[6] = (stdv != 0.f) ? coef * sc3 / (stdv * stdv * stdv) : 0.f;
        float alpha = n * (n + 1.f) * (n - 1.f) / ((n - 2.f) * (n - 3.f));
        float beta  = 3.f * (n - 1.f) * (n - 1.f) / ((n - 2.f) * (n - 3.f));
        op[7] = ((sc2 != 0.f) ? alpha * sc4 / (sc2 * sc2) : 0.f) - beta;
        op[8]  = sd / (n - 2.f);
        op[9]  = sd;
        op[10] = sad / (n - 2.f);
        op[11] = s2;
        op[12] = ma;
        op[13] = sad;
        op[14] = (sdd > 0.f) ? sqrtf(sdd) : 0.f;
        op[15] = cpos;
        op[16] = ccpos;
        op[17] = cntv[0]; op[18] = cntv[1]; op[19] = cntv[2];
        op[20] = cntv[3]; op[21] = cntv[4];
        op[22] = crx;
        op[23] = ccrx;
        op[24] = crov[5]; op[25] = crov[6]; op[26] = crov[7];   // q crossings
        op[27] = crov[0]; op[28] = crov[1]; op[29] = crov[2];   // tb crossings
        op[30] = crov[3]; op[31] = crov[4];
        op[32] = ampb[0];
        op[33] = am;  op[34] = av;  op[35] = ask; op[36] = aku;
        op[37] = rm;  op[38] = rv;  op[39] = rsk; op[40] = rku;

        // agg4 over the 9 autocorr lags
        const float n3 = 9.f;
        float acm = 0.f;
        for (int i = 0; i < 9; ++i) acm += ac_sh[i];
        acm /= n3;
        float b2 = 0.f, b3 = 0.f, b4 = 0.f, acmax = -__builtin_inff();
        for (int i = 0; i < 9; ++i) {
            float c = ac_sh[i] - acm, c2 = c * c;
            b2 += c2; b3 += c2 * c; b4 += c2 * c2;
            acmax = fmaxf(acmax, ac_sh[i]);
        }
        float acv_ = b2 / n3;
        float acsd = (acv_ > 0.f) ? sqrtf(acv_) : 0.f;
        float acs  = (acsd != 0.f) ? (n3 / ((n3 - 1.f) * (n3 - 2.f))) * b3 / (acsd * acsd * acsd) : 0.f;
        float alpha3 = n3 * (n3 + 1.f) * (n3 - 1.f) / ((n3 - 2.f) * (n3 - 3.f));
        float beta3  = 3.f * (n3 - 1.f) * (n3 - 1.f) / ((n3 - 2.f) * (n3 - 3.f));
        float ack  = ((b2 != 0.f) ? alpha3 * b4 / (b2 * b2) : 0.f) - beta3;
        op[41] = acm; op[42] = acv_; op[43] = acs; op[44] = ack;

        op[45] = q0; op[46] = q1; op[47] = q2;
        op[48] = tb0; op[49] = tb1; op[50] = tb2; op[51] = tb3; op[52] = tb4;

        float invmx = (acmax != 0.f) ? 1.f / acmax : 0.f;
        for (int i = 0; i < 9; ++i) {
            op[3128 + i] = ac_sh[i];
            op[3137 + i] = ac_sh[i] * invmx;
        }
    }
}

extern "C" void kernel_launch(void* const* d_in, const int* in_sizes, int n_in,
                              void* d_out, int out_size, void* d_ws, size_t ws_size,
                              hipStream_t stream) {
    (void)in_sizes; (void)n_in; (void)out_size; (void)d_ws; (void)ws_size;
    const float* x = (const float*)d_in[0];
    float* out = (float*)d_out;
    ts_features_kernel<<<dim3(BDIM * FDIM), dim3(NTH), 0, stream>>>(x, out);
}